// Trainer_3015067042012
// MI455X (gfx1250) — compile-verified
//
#include <hip/hip_runtime.h>
#include <cstdint>
#include <cstddef>

// Problem constants (match reference)
#define NRAYS 131072
#define NCC   64
#define NFF   128
#define NALL  192
constexpr float kNear = 1.0f;
constexpr float kFar  = 3.0f;
constexpr float kEpsW = 1e-5f;
constexpr float kEpsD = 1e-5f;

typedef __attribute__((ext_vector_type(2))) float v2f;
typedef __attribute__((ext_vector_type(8))) float v8f;

struct F3 { float x, y, z; };   // packed 12B (don't trust float3 layout)

// Compiler-only fence: LDS is in-order within a wave on CDNA5 (wave32 lockstep),
// we only need to stop clang from reordering/caching LDS accesses.
#define WFENCE() asm volatile("" ::: "memory")

__device__ __forceinline__ void ce(float& a, float& b, bool up) {
  float lo = fminf(a, b), hi = fmaxf(a, b);
  a = up ? lo : hi;
  b = up ? hi : lo;
}

// Cross-lane bitonic exchange: element index i = 4*lane + m, partner lane = lane^xl.
__device__ __forceinline__ void cx4(float* r, int xl, bool up, int lane) {
  bool lower = (lane & xl) == 0;
#pragma unroll
  for (int m = 0; m < 4; ++m) {
    float p = __shfl_xor(r[m], xl, 32);
    r[m] = (lower == up) ? fminf(r[m], p) : fmaxf(r[m], p);
  }
}

// In-lane tail of a bitonic merge (j=2 then j=1) on 4 registers.
__device__ __forceinline__ void inlane4(float* r, bool up) {
  ce(r[0], r[2], up); ce(r[1], r[3], up);
  ce(r[0], r[1], up); ce(r[2], r[3], up);
}

constexpr int WAVES   = 8;     // rays (waves) per 256-thread block
constexpr int WSTRIDE = 832;   // floats of LDS per wave slice (26.6 KB/block)

__global__ __launch_bounds__(256)
void nerf_resample_kernel(const float* __restrict__ ray_orig,
                          const float* __restrict__ ray_dir,
                          const float* __restrict__ sigma,
                          const float* __restrict__ t_noise,
                          const float* __restrict__ rand_var,
                          float* __restrict__ out)
{
  __shared__ float lds[WAVES * WSTRIDE];
  const int lane = threadIdx.x & 31;
  const int wid  = threadIdx.x >> 5;
  const int ray  = blockIdx.x * WAVES + wid;

  float* Z   = lds + wid * WSTRIDE; // [64]  coarse z_vals
  float* CDF = Z + 64;              // [65]
  float* MID = CDF + 72;            // [65]  z_mid bins
  float* ZF  = MID + 72;            // [128] sorted fine depths
  float* ZA  = ZF + 128;            // [192] merged sorted depths
  float* RV  = ZA + 192;            // [128] async-staged rand_var (16B aligned: 528f)

  // ---- Async stage rand_var -> LDS (overlaps all of Phase A) -------------
  // One global_load_async_to_lds_b128 per lane: lane copies bytes [16*lane, 16*lane+16).
  {
    const float* gsrc = rand_var + (size_t)ray * NFF + 4 * lane;
    uint32_t ldsoff = (uint32_t)(uintptr_t)(RV + 4 * lane);  // low 32 bits = LDS offset
    asm volatile("global_load_async_to_lds_b128 %0, %1, off"
                 :: "v"(ldsoff), "v"(gsrc) : "memory");
  }

  // ---------------- Phase A: coarse samples, CDF, bins ----------------
  const size_t rc = (size_t)ray * NCC;
  float tn0 = t_noise[rc + lane];
  float tn1 = t_noise[rc + 32 + lane];
  float sg0 = sigma[rc + lane];
  float sg1 = sigma[rc + 32 + lane];

  const float c63 = 1.0f / 63.0f, c64 = 1.0f / 64.0f;
  float z0 = kNear + (kFar - kNear) * ((float)lane        * c63 + tn0 * c64);
  float z1 = kNear + (kFar - kNear) * ((float)(lane + 32) * c63 + tn1 * c64);
  Z[lane] = z0; Z[lane + 32] = z1;
  WFENCE();
  float zp0 = (lane == 0) ? kNear : Z[lane - 1];   // z[c-1], prepend NEAR
  float zp1 = Z[lane + 31];

  float sdt0 = -sg0 * (z0 - zp0);
  float sdt1 = -sg1 * (z1 - zp1);

  // inclusive scan of sdt over the 64 coarse samples (two 32-wide chunks)
  float s0 = sdt0;
#pragma unroll
  for (int o = 1; o < 32; o <<= 1) { float n = __shfl_up(s0, o, 32); if (lane >= o) s0 += n; }
  float tot0 = __shfl(s0, 31, 32);
  float s1 = sdt1;
#pragma unroll
  for (int o = 1; o < 32; o <<= 1) { float n = __shfl_up(s1, o, 32); if (lane >= o) s1 += n; }
  s1 += tot0;

  // e[c] = exp(cumsum_incl).  weights telescope: sum_{c<j} w = 1 - e[j-1] + j*eps
  float e0 = __expf(s0), e1 = __expf(s1);
  float e63  = __shfl(e1, 31, 32);
  float wsum = (1.0f - e63) + 64.0f * kEpsW;
  float inv  = 1.0f / wsum;

  if (lane == 0) CDF[0] = 0.0f;
  CDF[lane + 1]  = (1.0f - e0 + (float)(lane + 1)  * kEpsW) * inv;
  CDF[lane + 33] = (1.0f - e1 + (float)(lane + 33) * kEpsW) * inv;

  if (lane == 0)  MID[0] = z0;                    // z[0]
  if (lane > 0)   MID[lane] = 0.5f * (zp0 + z0);  // mids 1..31
  MID[lane + 32] = 0.5f * (zp1 + z1);             // mids 32..63
  if (lane == 31) MID[64] = z1;                   // z[63]
  WFENCE();

  // ---- Wait for the async rand_var staging to land in LDS ----------------
#if __has_builtin(__builtin_amdgcn_s_wait_asynccnt)
  __builtin_amdgcn_s_wait_asynccnt(0);
#else
  asm volatile("s_wait_asynccnt 0" ::: "memory");
#endif
  WFENCE();

  // ---------------- Phase B: inverse-CDF fine sampling ----------------
  float zr[4];
#pragma unroll
  for (int i = 0; i < 4; ++i) {
    float v = RV[lane + 32 * i];
    int lo = 0, hi = 65;                          // searchsorted right on CDF[0..64]
#pragma unroll
    for (int it = 0; it < 7; ++it) {
      int mid = (lo + hi) >> 1;
      if (CDF[mid] <= v) lo = mid + 1; else hi = mid;
    }
    int below = lo - 1;                           // lo in [1,65]
    int above = (lo > 64) ? 64 : lo;
    float cg0 = CDF[below], cg1 = CDF[above];
    float b0  = MID[below], b1  = MID[above];
    float den = cg1 - cg0;
    den = (den < kEpsD) ? 1.0f : den;
    float tf = (v - cg0) / den;
    zr[i] = b0 + tf * (b1 - b0);
  }

  // ---------------- Phase C: in-register bitonic sort of 128 ----------------
  // element index i = 4*lane + m; direction = ((i & k) == 0)
  ce(zr[0], zr[1], true); ce(zr[2], zr[3], false);                     // k=2
  { bool up = (lane & 1)  == 0; inlane4(zr, up); }                     // k=4
  { bool up = (lane & 2)  == 0; cx4(zr, 1, up, lane); inlane4(zr, up); }                       // k=8
  { bool up = (lane & 4)  == 0; cx4(zr, 2, up, lane); cx4(zr, 1, up, lane); inlane4(zr, up); } // k=16
  { bool up = (lane & 8)  == 0; cx4(zr, 4, up, lane); cx4(zr, 2, up, lane);
    cx4(zr, 1, up, lane); inlane4(zr, up); }                                                   // k=32
  { bool up = (lane & 16) == 0; cx4(zr, 8, up, lane); cx4(zr, 4, up, lane);
    cx4(zr, 2, up, lane); cx4(zr, 1, up, lane); inlane4(zr, up); }                             // k=64
  { cx4(zr, 16, true, lane); cx4(zr, 8, true, lane); cx4(zr, 4, true, lane);
    cx4(zr, 2, true, lane); cx4(zr, 1, true, lane); inlane4(zr, true); }                       // k=128

#pragma unroll
  for (int m = 0; m < 4; ++m) ZF[4 * lane + m] = zr[m];
  WFENCE();

  // ---------------- Phase D: rank-merge coarse(64) + fine(128) ----------------
  // coarse z at position c + #{zf < z}; fine zf at position f + #{zv <= zf}
  {
    int lo = 0, hi = 128;
#pragma unroll
    for (int it = 0; it < 7; ++it) { int mid = (lo + hi) >> 1; if (ZF[mid] < z0) lo = mid + 1; else hi = mid; }
    ZA[lane + lo] = z0;
    lo = 0; hi = 128;
#pragma unroll
    for (int it = 0; it < 7; ++it) { int mid = (lo + hi) >> 1; if (ZF[mid] < z1) lo = mid + 1; else hi = mid; }
    ZA[lane + 32 + lo] = z1;
  }
#pragma unroll
  for (int m = 0; m < 4; ++m) {
    float v = zr[m];
    int lo = 0, hi = 64;
#pragma unroll
    for (int it = 0; it < 6; ++it) { int mid = (lo + hi) >> 1; if (Z[mid] <= v) lo = mid + 1; else hi = mid; }
    ZA[4 * lane + m + lo] = v;
  }
  WFENCE();

  // ---------------- Phase E: outputs ----------------
  const float* ro = ray_orig + (size_t)ray * 3;
  const float* rd = ray_dir  + (size_t)ray * 3;
  float ox = ro[0], oy = ro[1], oz = ro[2];
  float dx = rd[0], dy = rd[1], dz = rd[2];

  float* coords = out;
  float* vdir   = out + (size_t)NRAYS * NALL * 3;
  float* depth  = out + (size_t)NRAYS * NALL * 3 * 2;
  float* deltas = depth + (size_t)NRAYS * NALL;

  const size_t o3 = (size_t)ray * NALL * 3;   // per-ray offset into coords/vdir
  const size_t o1 = (size_t)ray * NALL;       // per-ray offset into depth/deltas

#pragma unroll
  for (int i = 0; i < 6; ++i) {
    int s = 32 * i + lane;
    float z  = ZA[s];
    float zp = (s == 0) ? kNear : ZA[s - 1];
    __builtin_nontemporal_store(z,      depth  + o1 + s);   // streaming outputs:
    __builtin_nontemporal_store(z - zp, deltas + o1 + s);   // keep them out of L2
    F3 d3{dx, dy, dz};
    *(F3*)(vdir + o3 + (size_t)s * 3) = d3;
  }

#if __has_builtin(__builtin_amdgcn_wmma_f32_16x16x4_f32)
  // coords = dir (16x4, col K=0) x z_row (4x16, row K=0) + orig  via V_WMMA_F32_16X16X4_F32.
  // D rows 0..2 = x/y/z for 16 samples; lanes 0..15 each hold one sample's xyz.
  v2f A;
  A.x = (lane == 0) ? dx : ((lane == 1) ? dy : ((lane == 2) ? dz : 0.0f)); // K=0 (lanes<16), K=2 zero
  A.y = 0.0f;                                                              // K=1,3 zero
  const bool lowhalf = (lane < 16);
  v8f C = {0.f, 0.f, 0.f, 0.f, 0.f, 0.f, 0.f, 0.f};
  C[0] = lowhalf ? ox : 0.0f;   // row M=0 (x), lanes16-31 are row 8 -> 0
  C[1] = lowhalf ? oy : 0.0f;
  C[2] = lowhalf ? oz : 0.0f;
#pragma unroll
  for (int t = 0; t < 12; ++t) {
    v2f B;
    B.x = lowhalf ? ZA[16 * t + lane] : 0.0f;   // row K=0 = depths; row K=2 zero
    B.y = 0.0f;                                 // rows K=1,3 zero
    v8f D = __builtin_amdgcn_wmma_f32_16x16x4_f32(false, A, false, B,
                                                  (short)0, C, false, false);
    if (lowhalf) {
      F3 p{D[0], D[1], D[2]};
      *(F3*)(coords + o3 + (size_t)(16 * t + lane) * 3) = p;
    }
  }
#else
#pragma unroll
  for (int i = 0; i < 6; ++i) {
    int s = 32 * i + lane;
    float z = ZA[s];
    F3 p{ox + dx * z, oy + dy * z, oz + dz * z};
    *(F3*)(coords + o3 + (size_t)s * 3) = p;
  }
#endif
}

extern "C" void kernel_launch(void* const* d_in, const int* in_sizes, int n_in,
                              void* d_out, int out_size, void* d_ws, size_t ws_size,
                              hipStream_t stream) {
  (void)in_sizes; (void)n_in; (void)out_size; (void)d_ws; (void)ws_size;
  const float* ray_orig = (const float*)d_in[0];
  const float* ray_dir  = (const float*)d_in[1];
  const float* sigma    = (const float*)d_in[2];
  const float* t_noise  = (const float*)d_in[3];
  const float* rand_var = (const float*)d_in[4];
  float* out = (float*)d_out;

  dim3 grid(NRAYS / WAVES);   // 16384 blocks, 8 rays each
  dim3 block(256);            // 8 wave32 per block (one ray per wave)
  hipLaunchKernelGGL(nerf_resample_kernel, grid, block, 0, stream,
                     ray_orig, ray_dir, sigma, t_noise, rand_var, out);
}